// RNNLM_76510547411091
// MI455X (gfx1250) — compile-verified
//
#include <hip/hip_runtime.h>
#include <hip/hip_bf16.h>

#define VOCAB 32000
#define EMB   32
#define REC   16
#define SEQ   256
#define BATCH 32
#define ROWS  (SEQ * BATCH)   // 8192
#define ROWT  (ROWS / 16)     // 512 row tiles
#define COLT  (VOCAB / 16)    // 2000 col tiles

typedef float v2f __attribute__((ext_vector_type(2)));
typedef float v8f __attribute__((ext_vector_type(8)));

// D = A(16x4 f32) x B(4x16 f32) + C(16x16 f32), wave32.
__device__ __forceinline__ v8f wmma4(v2f a, v2f b, v8f c) {
    return __builtin_amdgcn_wmma_f32_16x16x4_f32(
        /*neg_a=*/false, a, /*neg_b=*/false, b,
        /*c_mod=*/(short)0, c, /*reuse_a=*/false, /*reuse_b=*/false);
}

// ---------------------------------------------------------------------------
// K1: ih[row][r] = bias1[r] + dot(emb[token[row]], U[r])   (row = t*BATCH + b)
// ---------------------------------------------------------------------------
__global__ void __launch_bounds__(256) k1_ih(const int* __restrict__ tok,
                                             const float* __restrict__ emb,
                                             const float* __restrict__ U,
                                             const float* __restrict__ b1,
                                             float* __restrict__ ih) {
    int tid = blockIdx.x * 256 + threadIdx.x;   // 0 .. ROWS*REC-1
    int row = tid >> 4;
    int r   = tid & 15;
    int t   = tok[row];
    const float4* e = (const float4*)(emb + (long)t * EMB);  // 128B aligned
    const float4* u = (const float4*)(U + r * EMB);
    float acc = b1[r];
#pragma unroll
    for (int k = 0; k < 8; ++k) {
        float4 ev = e[k], uv = u[k];
        acc += ev.x * uv.x + ev.y * uv.y + ev.z * uv.z + ev.w * uv.w;
    }
    ih[row * REC + r] = acc;
}

// ---------------------------------------------------------------------------
// K2: sequential recurrence, one wave per batch chain. hidden[t] = h_{t-1}.
// ---------------------------------------------------------------------------
__global__ void __launch_bounds__(32) k2_rnn(const float* __restrict__ ih,
                                             const float* __restrict__ W,
                                             const float* __restrict__ b2,
                                             const float* __restrict__ h0,
                                             float* __restrict__ hidden) {
    const int b    = blockIdx.x;        // batch chain
    const int lane = threadIdx.x;       // 0..31
    const int r    = lane & 15;
    float w[REC];
#pragma unroll
    for (int j = 0; j < REC; ++j) w[j] = W[r * REC + j];
    const float bias = b2[r];
    float h = h0[r];
    // software pipeline, prefetch depth 2 (ih addresses are data-independent)
    float p0 = ih[(0 * BATCH + b) * REC + r];
    float p1 = ih[(1 * BATCH + b) * REC + r];
    for (int t = 0; t < SEQ; ++t) {
        float p2 = (t + 2 < SEQ) ? ih[((t + 2) * BATCH + b) * REC + r] : 0.f;
        if (lane < 16) hidden[(t * BATCH + b) * REC + r] = h;   // pre-update h
        float acc = p0 + bias;
#pragma unroll
        for (int j = 0; j < REC; ++j) acc += w[j] * __shfl(h, j, 32);
        h = tanhf(acc);
        p0 = p1; p1 = p2;
    }
}

// ---------------------------------------------------------------------------
// K3: per-row logsumexp stats over VOCAB via WMMA sweep (online softmax).
//     One block per 16-row tile; wave wv handles col tiles wv, wv+8, ...
// ---------------------------------------------------------------------------
__global__ void __launch_bounds__(256) k3_stats(const float* __restrict__ hidden,
                                                const float* __restrict__ V,
                                                float* __restrict__ Z) {
    __shared__ float sm[8][16];
    __shared__ float ss[8][16];
    const int lane = threadIdx.x & 31;
    const int wv   = threadIdx.x >> 5;
    const int half = lane >> 4;          // 0: lanes 0-15, 1: lanes 16-31
    const int nm   = lane & 15;
    const int rowbase = blockIdx.x * 16;

    // hoist A tile (hidden rows), v2f = K pair {2*half+j} per kk chunk
    v2f a[4];
    const float* hrow = hidden + (rowbase + nm) * REC + 2 * half;
#pragma unroll
    for (int kk = 0; kk < 4; ++kk) a[kk] = *(const v2f*)(hrow + 4 * kk);

    float m[8], s[8];
#pragma unroll
    for (int r = 0; r < 8; ++r) { m[r] = -INFINITY; s[r] = 0.f; }

    for (int ct = wv; ct < COLT; ct += 8) {
        const float* vrow = V + (ct * 16 + nm) * REC + 2 * half;
        v8f c = {0.f, 0.f, 0.f, 0.f, 0.f, 0.f, 0.f, 0.f};
#pragma unroll
        for (int kk = 0; kk < 4; ++kk) {
            v2f bv = *(const v2f*)(vrow + 4 * kk);
            c = wmma4(a[kk], bv, c);
        }
        // online logsumexp per lane-slot: VGPR r = row (r + 8*half)
#pragma unroll
        for (int r = 0; r < 8; ++r) {
            float d = c[r];
            float M = fmaxf(m[r], d);
            s[r] = s[r] * __expf(m[r] - M) + __expf(d - M);
            m[r] = M;
        }
    }
    // reduce across the 16 lanes of each half (xor masks < 16 stay in-half)
#pragma unroll
    for (int mask = 1; mask < 16; mask <<= 1) {
#pragma unroll
        for (int r = 0; r < 8; ++r) {
            float om = __shfl_xor(m[r], mask, 32);
            float os = __shfl_xor(s[r], mask, 32);
            float M  = fmaxf(m[r], om);
            s[r] = s[r] * __expf(m[r] - M) + os * __expf(om - M);
            m[r] = M;
        }
    }
    if (nm == 0) {  // lanes 0 and 16 publish their 8 rows
#pragma unroll
        for (int r = 0; r < 8; ++r) {
            sm[wv][8 * half + r] = m[r];
            ss[wv][8 * half + r] = s[r];
        }
    }
    __syncthreads();
    if (threadIdx.x < 16) {
        int j = threadIdx.x;
        float M = sm[0][j], S = ss[0][j];
#pragma unroll
        for (int w = 1; w < 8; ++w) {
            float om = sm[w][j], os = ss[w][j];
            float Mn = fmaxf(M, om);
            S = S * __expf(M - Mn) + os * __expf(om - Mn);
            M = Mn;
        }
        Z[rowbase + j] = M + __logf(S);
    }
}

// ---------------------------------------------------------------------------
// K4: recompute logits via WMMA, write out = logit - Z (the 1 GB stream).
//     grid (512, 25); block covers 80 col tiles, 10 per wave, contiguous.
//     Output is write-once / never re-read on device -> non-temporal stores
//     (TH=NT) so the stream drains to HBM without evicting V from L2.
// ---------------------------------------------------------------------------
__global__ void __launch_bounds__(256) k4_write(const float* __restrict__ hidden,
                                                const float* __restrict__ V,
                                                const float* __restrict__ Z,
                                                float* __restrict__ out) {
    const int lane = threadIdx.x & 31;
    const int wv   = threadIdx.x >> 5;
    const int half = lane >> 4;
    const int nm   = lane & 15;
    const int rowbase = blockIdx.x * 16;

    v2f a[4];
    const float* hrow = hidden + (rowbase + nm) * REC + 2 * half;
#pragma unroll
    for (int kk = 0; kk < 4; ++kk) a[kk] = *(const v2f*)(hrow + 4 * kk);

    float z[8];
#pragma unroll
    for (int r = 0; r < 8; ++r) z[r] = Z[rowbase + 8 * half + r];

    const int ct0 = blockIdx.y * 80 + wv * 10;
    for (int i = 0; i < 10; ++i) {
        int ct = ct0 + i;
        const float* vrow = V + (ct * 16 + nm) * REC + 2 * half;
        v8f c = {0.f, 0.f, 0.f, 0.f, 0.f, 0.f, 0.f, 0.f};
#pragma unroll
        for (int kk = 0; kk < 4; ++kk) {
            v2f bv = *(const v2f*)(vrow + 4 * kk);
            c = wmma4(a[kk], bv, c);
        }
        float* obase = out + (long)(rowbase + 8 * half) * VOCAB + ct * 16 + nm;
#pragma unroll
        for (int r = 0; r < 8; ++r)
            __builtin_nontemporal_store(c[r] - z[r], obase + (long)r * VOCAB);
    }
}

// ---------------------------------------------------------------------------
extern "C" void kernel_launch(void* const* d_in, const int* in_sizes, int n_in,
                              void* d_out, int out_size, void* d_ws, size_t ws_size,
                              hipStream_t stream) {
    const int*   tok = (const int*)d_in[0];      // [SEQ, BATCH]
    const float* emb = (const float*)d_in[1];    // [VOCAB, EMB]
    const float* U   = (const float*)d_in[2];    // [REC, EMB]
    const float* W   = (const float*)d_in[3];    // [REC, REC]
    const float* V   = (const float*)d_in[4];    // [VOCAB, REC]
    const float* b1  = (const float*)d_in[5];    // [REC]
    const float* b2  = (const float*)d_in[6];    // [REC]
    const float* h0  = (const float*)d_in[7];    // [REC]
    float* out = (float*)d_out;                  // [SEQ, BATCH, VOCAB] f32

    char* ws = (char*)d_ws;
    float* hidden = (float*)(ws);                              // 8192*16 f32
    float* ihbuf  = (float*)(ws + (size_t)ROWS * REC * 4);     // 8192*16 f32
    float* Zbuf   = (float*)(ws + (size_t)2 * ROWS * REC * 4); // 8192 f32

    k1_ih  <<<(ROWS * REC) / 256, 256, 0, stream>>>(tok, emb, U, b1, ihbuf);
    k2_rnn <<<BATCH, 32, 0, stream>>>(ihbuf, W, b2, h0, hidden);
    k3_stats<<<ROWT, 256, 0, stream>>>(hidden, V, Zbuf);
    k4_write<<<dim3(ROWT, 25), 256, 0, stream>>>(hidden, V, Zbuf, out);
}